// NodePredictionGCN_88424786690105
// MI455X (gfx1250) — compile-verified
//
#include <hip/hip_runtime.h>

#define NN 50000
#define NE 640000
#define INF 128
#define HIDF 256
#define OUTF 128

typedef __attribute__((ext_vector_type(2))) float v2f;
typedef __attribute__((ext_vector_type(8))) float v8f;
typedef __attribute__((ext_vector_type(4))) unsigned int u32x4;
typedef __attribute__((ext_vector_type(8))) int i32x8;
typedef __attribute__((ext_vector_type(4))) int i32x4;

// ---------------------------------------------------------------- utilities
__global__ void zero_f32(float* __restrict__ p, long n) {
  long i = blockIdx.x * (long)blockDim.x + threadIdx.x;
  if (i < n) p[i] = 0.0f;
}

// deg[d] += 1 for each edge destination (self-loop folded in later as +1)
__global__ void degree_kernel(const int* __restrict__ dst, float* __restrict__ deg, int E) {
  int i = blockIdx.x * blockDim.x + threadIdx.x;
  if (i < E)
    __hip_atomic_fetch_add(&deg[dst[i]], 1.0f, __ATOMIC_RELAXED, __HIP_MEMORY_SCOPE_AGENT);
}

// deg -> dinv = rsqrt(deg + 1)   (deg+1 >= 1, so no zero guard needed)
__global__ void dinv_kernel(float* __restrict__ deg, int n) {
  int i = blockIdx.x * blockDim.x + threadIdx.x;
  if (i < n) deg[i] = rsqrtf(deg[i] + 1.0f);
}

// --------------------------------------------------- TDM 2D tile -> LDS
// Pack a Tensor DMA Descriptor (D#) per CDNA5 ISA §8.3/§8.4 for a 2D fp32 tile:
//   tile = k_rows x tile_cols elements, row-major, from a tensor with row
//   length n_total (elements) and row stride n_total, into LDS at lds_byte
//   (rows stored contiguously: tile_cols*4 bytes per row).
#if __has_builtin(__builtin_amdgcn_tensor_load_to_lds)
#define HAVE_TDM 1
__device__ __forceinline__ void tdm_load_tile_2d(unsigned lds_byte, const float* gptr,
                                                 unsigned n_total, unsigned k_rows,
                                                 unsigned tile_cols) {
  unsigned long long ga = (unsigned long long)(uintptr_t)gptr;
  u32x4 g0;
  g0.x = 1u;                                             // count=1, user mode
  g0.y = lds_byte;                                       // lds_addr [63:32]
  g0.z = (unsigned)(ga & 0xFFFFFFFFu);                   // global_addr[31:0]
  g0.w = (unsigned)((ga >> 32) & 0x01FFFFFFu) | (2u << 30); // global_addr[56:32], type=2
  i32x8 g1;
  g1[0] = 0x00020000;                                    // wg_mask=0, data_size=2 (4B)
  g1[1] = (int)((n_total & 0xFFFFu) << 16);              // tensor_dim0[15:0] @ bits63:48
  g1[2] = (int)((n_total >> 16) | ((k_rows & 0xFFFFu) << 16));  // dim0 hi | tensor_dim1 lo
  g1[3] = (int)((k_rows >> 16) | (tile_cols << 16));     // dim1 hi | tile_dim0
  g1[4] = (int)(k_rows & 0xFFFFu);                       // tile_dim1 = k_rows, tile_dim2=0
  g1[5] = (int)n_total;                                  // tensor_dim0_stride[31:0]
  g1[6] = 0;                                             // stride0 hi | stride1 lo
  g1[7] = 0;                                             // stride1 hi (unused, 2D)
  i32x4 z4 = {0, 0, 0, 0};
  i32x8 z8 = {0, 0, 0, 0, 0, 0, 0, 0};
  // 6-arg form (this toolchain): (g0, g1, g2, g3, extra, cpol)
  __builtin_amdgcn_tensor_load_to_lds(g0, g1, z4, z4, z8, 0);
}
#else
#define HAVE_TDM 0
#endif

// ------------------------------------------------------- fp32 WMMA GEMM
// C[M,N] = A[M,K] x B[K,N], fp32 via V_WMMA_F32_16X16X4_F32.
// grid = (ceil(M/16/8), N/64); block = 256 (8 waves).
// All 8 waves of a block share one 64-wide N panel of B, staged in LDS by the
// Tensor Data Mover; each wave computes a 16(M) x 64(N) tile (4 accumulators,
// A fragment reused 4x, B fragments served from LDS at ds latency).
template <int K>
__global__ __launch_bounds__(256)
void gemm_f32_wmma(const float* __restrict__ A, const float* __restrict__ B,
                   float* __restrict__ C, int M, int N) {
  __shared__ float Bs[K * 64];                 // 32 KB (K=128) / 64 KB (K=256)
  const int tn = blockIdx.y << 6;

  // ---- stage B[:, tn:tn+64] into LDS
#if HAVE_TDM
  if (threadIdx.x == 0) {
    unsigned lds_byte = (unsigned)(uintptr_t)&Bs[0];  // low 32 bits of LDS aperture addr
    tdm_load_tile_2d(lds_byte, B + tn, (unsigned)N, (unsigned)K, 64u);
    __builtin_amdgcn_s_wait_tensorcnt(0);
  }
#else
  for (int idx = threadIdx.x; idx < K * 16; idx += 256) {
    int r = idx >> 4, c4 = (idx & 15) << 2;
    *(float4*)&Bs[r * 64 + c4] = *(const float4*)(B + (long)r * N + tn + c4);
  }
#endif
  __syncthreads();

  const int tilesM = M >> 4;
  const int tm = blockIdx.x * 8 + (threadIdx.x >> 5);
  if (tm >= tilesM) return;                    // wave-uniform; after barrier

  const int lane = threadIdx.x & 31;
  const int l = lane & 15;                     // N / M index within half-wave
  const int h = lane >> 4;                     // half-wave select

  // A fragment per 16x16x4 step: lane(l,h) holds A[tm*16+l][k0+2h .. k0+2h+1]
  const float* arow = A + (long)(tm * 16 + l) * K + 2 * h;
  const float* bs0 = &Bs[l];                   // + (k)*64 per row

  v8f acc0 = {}, acc1 = {}, acc2 = {}, acc3 = {};
#pragma unroll 4
  for (int k0 = 0; k0 < K; k0 += 4) {
    v2f a;
    a.x = arow[k0];
    a.y = arow[k0 + 1];
    const float* brow0 = bs0 + (k0 + 2 * h) * 64;   // B[k0+2h][tn+l+..]
    const float* brow1 = brow0 + 64;                 // B[k0+2h+1][..]
    v2f b;
    b.x = brow0[0];  b.y = brow1[0];
    acc0 = __builtin_amdgcn_wmma_f32_16x16x4_f32(false, a, false, b, (short)0, acc0, false, false);
    b.x = brow0[16]; b.y = brow1[16];
    acc1 = __builtin_amdgcn_wmma_f32_16x16x4_f32(false, a, false, b, (short)0, acc1, false, false);
    b.x = brow0[32]; b.y = brow1[32];
    acc2 = __builtin_amdgcn_wmma_f32_16x16x4_f32(false, a, false, b, (short)0, acc2, false, false);
    b.x = brow0[48]; b.y = brow1[48];
    acc3 = __builtin_amdgcn_wmma_f32_16x16x4_f32(false, a, false, b, (short)0, acc3, false, false);
  }

  // C/D layout: VGPR j -> row M = 8*h + j, col N = l
  float* crow = C + (long)(tm * 16 + 8 * h) * N + tn + l;
#pragma unroll
  for (int j = 0; j < 8; ++j) {
    crow[(long)j * N +  0] = acc0[j];
    crow[(long)j * N + 16] = acc1[j];
    crow[(long)j * N + 32] = acc2[j];
    crow[(long)j * N + 48] = acc3[j];
  }
}

// ------------------------------------------------------- edge scatter-add
// agg[dst] += dinv[src]*dinv[dst] * h[src]; F/4 threads per edge, float4 gathers,
// near-memory f32 atomics (no return -> non-returning atomic, STOREcnt path).
template <int F>
__global__ __launch_bounds__(256)
void scatter_kernel(const float* __restrict__ h, const int* __restrict__ src,
                    const int* __restrict__ dst, const float* __restrict__ dinv,
                    float* __restrict__ agg, int E) {
  constexpr int TPE = F / 4;
  long gid = blockIdx.x * (long)blockDim.x + threadIdx.x;
  int e = (int)(gid / TPE);
  if (e >= E) return;
  int f = (int)(gid % TPE) * 4;
  int s = src[e], d = dst[e];
  float nrm = dinv[s] * dinv[d];
  float4 v = *(const float4*)(h + (long)s * F + f);
  float* o = agg + (long)d * F + f;
  __hip_atomic_fetch_add(o + 0, nrm * v.x, __ATOMIC_RELAXED, __HIP_MEMORY_SCOPE_AGENT);
  __hip_atomic_fetch_add(o + 1, nrm * v.y, __ATOMIC_RELAXED, __HIP_MEMORY_SCOPE_AGENT);
  __hip_atomic_fetch_add(o + 2, nrm * v.z, __ATOMIC_RELAXED, __HIP_MEMORY_SCOPE_AGENT);
  __hip_atomic_fetch_add(o + 3, nrm * v.w, __ATOMIC_RELAXED, __HIP_MEMORY_SCOPE_AGENT);
}

// -------------------------------- self-loop + bias (+ optional ReLU), in place OK
__global__ void bias_self_act(const float* __restrict__ agg, const float* __restrict__ hh,
                              const float* __restrict__ bias, const float* __restrict__ dinv,
                              float* __restrict__ y, long total, int F, int do_relu) {
  long gid = blockIdx.x * (long)blockDim.x + threadIdx.x;
  if (gid >= total) return;
  int i = (int)(gid / F);
  int f = (int)(gid % F);
  float di = dinv[i];
  float v = agg[gid] + di * di * hh[gid] + bias[f];
  y[gid] = do_relu ? fmaxf(v, 0.0f) : v;
}

// ---------------------------------------------------------------- launch
extern "C" void kernel_launch(void* const* d_in, const int* in_sizes, int n_in,
                              void* d_out, int out_size, void* d_ws, size_t ws_size,
                              hipStream_t stream) {
  const float* x  = (const float*)d_in[0];
  const int*   ei = (const int*)d_in[1];
  const int*   src = ei;            // edge_index[0]
  const int*   dst = ei + NE;       // edge_index[1]
  const float* W1 = (const float*)d_in[2];
  const float* b1 = (const float*)d_in[3];
  const float* W2 = (const float*)d_in[4];
  const float* b2 = (const float*)d_in[5];
  float* out = (float*)d_out;

  float* ws   = (float*)d_ws;
  float* dinv = ws;                          // NN
  float* h1   = dinv + NN;                   // NN*HIDF (later reused as y1 in place)
  float* agg1 = h1 + (long)NN * HIDF;        // NN*HIDF
  float* h2   = agg1 + (long)NN * HIDF;      // NN*OUTF

  const long nh1 = (long)NN * HIDF;          // 12.8M
  const long nh2 = (long)NN * OUTF;          // 6.4M

  // --- zero accumulation targets (atomics accumulate; must re-zero every call)
  zero_f32<<<(NN + 255) / 256, 256, 0, stream>>>(dinv, NN);
  zero_f32<<<(int)((nh1 + 255) / 256), 256, 0, stream>>>(agg1, nh1);
  zero_f32<<<(int)((nh2 + 255) / 256), 256, 0, stream>>>(out, nh2);

  // --- symmetric norm: deg -> dinv
  degree_kernel<<<(NE + 255) / 256, 256, 0, stream>>>(dst, dinv, NE);
  dinv_kernel<<<(NN + 255) / 256, 256, 0, stream>>>(dinv, NN);

  const int tilesM = NN / 16;                       // 3125
  const int gx = (tilesM + 7) / 8;                  // 391 blocks of 8 waves

  // --- layer 1: h1 = x @ W1  (M=50000, K=128, N=256)
  gemm_f32_wmma<INF><<<dim3(gx, HIDF / 64), 256, 0, stream>>>(x, W1, h1, NN, HIDF);
  // edge aggregation into agg1
  {
    long thr = (long)NE * (HIDF / 4);
    scatter_kernel<HIDF><<<(int)((thr + 255) / 256), 256, 0, stream>>>(h1, src, dst, dinv, agg1, NE);
  }
  // y1 = relu(agg1 + dinv^2*h1 + b1), written over h1 (element-wise, in-place safe)
  bias_self_act<<<(int)((nh1 + 255) / 256), 256, 0, stream>>>(agg1, h1, b1, dinv, h1, nh1, HIDF, 1);

  // --- layer 2: h2 = y1 @ W2  (M=50000, K=256, N=128)
  gemm_f32_wmma<HIDF><<<dim3(gx, OUTF / 64), 256, 0, stream>>>(h1, W2, h2, NN, OUTF);
  // edge aggregation straight into d_out
  {
    long thr = (long)NE * (OUTF / 4);
    scatter_kernel<OUTF><<<(int)((thr + 255) / 256), 256, 0, stream>>>(h2, src, dst, dinv, out, NE);
  }
  // out = out + dinv^2*h2 + b2 (no relu), in place
  bias_self_act<<<(int)((nh2 + 255) / 256), 256, 0, stream>>>(out, h2, b2, dinv, out, nh2, OUTF, 0);
}